// Top1Gate_60610578481609
// MI455X (gfx1250) — compile-verified
//
#include <hip/hip_runtime.h>
#include <hip/hip_bf16.h>

#define SDIM 8192
#define EDIM 64
#define DDIM 4096
#define CAP  128
static constexpr long SEC = (long)SDIM * EDIM * CAP;   // 67,108,864

typedef float v2f __attribute__((ext_vector_type(2)));
typedef float v4f __attribute__((ext_vector_type(4)));
typedef float v8f __attribute__((ext_vector_type(8)));

// ---------------------------------------------------------------------------
// 1) Zero-fill the 512MB output with non-temporal 128-bit stores; also zero
//    the me_sum accumulator in workspace.
// ---------------------------------------------------------------------------
__global__ __launch_bounds__(256)
void zero_kernel(float* __restrict__ out, float* __restrict__ me_sum)
{
    const long n4 = (2 * SEC) / 4;                    // 33,554,432 float4s
    const long stride = (long)gridDim.x * blockDim.x;
    const long gid = (long)blockIdx.x * blockDim.x + threadIdx.x;
    v4f z = {0.f, 0.f, 0.f, 0.f};
    for (long i = gid; i < n4; i += stride)
        __builtin_nontemporal_store(z, (v4f*)out + i);
    if (gid == 0) out[2 * SEC] = 0.f;                 // last scalar element
    if (gid < EDIM) me_sum[gid] = 0.f;
}

// ---------------------------------------------------------------------------
// 2) Fused GEMM (logits = x @ wg^T via V_WMMA_F32_16X16X4_F32) + softmax/
//    argmax/top-gate + column-sum accumulation for the aux loss.
//    Block: 256 threads = 8 waves; block computes 32 rows x 64 experts.
//    Wave (wm,wn) computes the 16x16 tile at rows m0+wm*16, cols wn*16.
// ---------------------------------------------------------------------------
__global__ __launch_bounds__(256)
void gemm_softmax_kernel(const float* __restrict__ x,
                         const float* __restrict__ wg,
                         float* __restrict__ gate_top,
                         int*   __restrict__ idx_out,
                         float* __restrict__ me_sum)
{
    __shared__ float lds[32][65];                     // padded: no bank conflicts

    const int tid  = threadIdx.x;
    const int wave = tid >> 5;
    const int lane = tid & 31;
    const int r    = lane & 15;                       // M (for A) / N (for B)
    const int kh   = lane >> 4;                       // K half-select
    const int m0   = blockIdx.x * 32;
    const int wm   = wave >> 2;                       // 0..1
    const int wn   = wave & 3;                        // 0..3

    // A 16x4 f32 layout: lane(M)=r, VGPR0/1 hold K = kh*2, kh*2+1
    const float* aptr = x  + (long)(m0 + wm * 16 + r) * DDIM + kh * 2;
    // B 4x16 f32 layout (K x N): lane(N)=r, VGPR0/1 hold K = kh*2, kh*2+1
    // B[k][n] = wg[n][k]  (wg is [E, D] row-major)
    const float* bptr = wg + (long)(wn * 16 + r) * DDIM + kh * 2;

    v8f c = {};
    for (int k = 0; k < DDIM; k += 16) {
#pragma unroll
        for (int u = 0; u < 4; ++u) {
            v2f a = *(const v2f*)(aptr + k + u * 4);
            v2f b = *(const v2f*)(bptr + k + u * 4);
            c = __builtin_amdgcn_wmma_f32_16x16x4_f32(
                    /*neg_a=*/false, a, /*neg_b=*/false, b,
                    /*c_mod=*/(short)0, c, /*reuse_a=*/false, /*reuse_b=*/false);
        }
    }

    // C/D layout: VGPR i -> M = i (lanes 0-15) or 8+i (lanes 16-31), N = lane&15
    const int mloc = wm * 16 + kh * 8;
    const int col  = wn * 16 + r;
#pragma unroll
    for (int i = 0; i < 8; ++i)
        lds[mloc + i][col] = c[i];
    __syncthreads();

    // Softmax / argmax / top gate: thread t handles local row t (32 rows)
    if (tid < 32) {
        const int srow = m0 + tid;
        float mx = lds[tid][0];
        int   arg = 0;
        for (int e = 1; e < EDIM; ++e) {
            float v = lds[tid][e];
            if (v > mx) { mx = v; arg = e; }          // first-max: matches argmax
        }
        float sum = 0.f;
        for (int e = 0; e < EDIM; ++e) sum += expf(lds[tid][e] - mx);
        const float inv = 1.f / sum;
        for (int e = 0; e < EDIM; ++e)
            lds[tid][e] = expf(lds[tid][e] - mx) * inv;   // overwrite with probs
        gate_top[srow] = inv;                          // exp(mx-mx)*inv
        idx_out[srow]  = arg;
    }
    __syncthreads();

    // Column sums of the 32x64 prob tile -> accumulate into global me_sum
    if (tid < EDIM) {
        float cs = 0.f;
#pragma unroll
        for (int t2 = 0; t2 < 32; ++t2) cs += lds[t2][tid];
        atomicAdd(&me_sum[tid], cs);
    }
}

// ---------------------------------------------------------------------------
// 3) Per-chunk (128 tokens) expert histogram + ordered local rank.
// ---------------------------------------------------------------------------
__global__ __launch_bounds__(128)
void rank_kernel(const int* __restrict__ idx,
                 int* __restrict__ lrank,
                 int* __restrict__ hist_g)
{
    __shared__ int lidx[128];
    __shared__ int hist[EDIM];
    const int t = threadIdx.x;
    const int cblk = blockIdx.x;
    if (t < EDIM) hist[t] = 0;
    __syncthreads();
    const int myidx = idx[cblk * 128 + t];
    lidx[t] = myidx;
    atomicAdd(&hist[myidx], 1);
    __syncthreads();
    int cnt = 0;
    for (int j = 0; j < t; ++j) cnt += (lidx[j] == myidx);
    lrank[cblk * 128 + t] = cnt;
    if (t < EDIM) hist_g[cblk * EDIM + t] = hist[t];
}

// ---------------------------------------------------------------------------
// 4) Per-expert exclusive prefix over 64 chunks -> base offsets; ce counts
//    fall out of the running totals; finalize l_aux into d_out[0].
// ---------------------------------------------------------------------------
__global__ __launch_bounds__(64)
void prefix_laux_kernel(const int* __restrict__ hist_g,
                        int* __restrict__ base_g,
                        const float* __restrict__ me_sum,
                        float* __restrict__ out0)
{
    __shared__ float red[EDIM];
    const int e = threadIdx.x;
    int run = 0;
    for (int cb = 0; cb < SDIM / 128; ++cb) {
        base_g[cb * EDIM + e] = run;
        run += hist_g[cb * EDIM + e];
    }
    red[e] = me_sum[e] * (float)run;                  // me_sum * ce_count
    __syncthreads();
    for (int off = 32; off > 0; off >>= 1) {
        if (e < off) red[e] += red[e + off];
        __syncthreads();
    }
    // l_aux = E * sum(me_sum*ce_cnt) / S^2
    if (e == 0) out0[0] = (float)EDIM * red[0] / ((float)SDIM * (float)SDIM);
}

// ---------------------------------------------------------------------------
// 5) Scatter: one nonzero per surviving token into combine + dispatch.
// ---------------------------------------------------------------------------
__global__ __launch_bounds__(256)
void scatter_kernel(const int* __restrict__ idx,
                    const int* __restrict__ lrank,
                    const int* __restrict__ base_g,
                    const float* __restrict__ gate_top,
                    float* __restrict__ out)
{
    const int s = blockIdx.x * 256 + threadIdx.x;
    if (s >= SDIM) return;
    const int e   = idx[s];
    const int loc = base_g[(s >> 7) * EDIM + e] + lrank[s];
    if (loc < CAP) {
        const long off = 1 + (long)s * EDIM * CAP + (long)e * CAP + loc;
        out[off]       = gate_top[s];                 // combine1_sec
        out[off + SEC] = 1.0f;                        // dispatch_mask (as float)
    }
}

// ---------------------------------------------------------------------------
extern "C" void kernel_launch(void* const* d_in, const int* in_sizes, int n_in,
                              void* d_out, int out_size, void* d_ws, size_t ws_size,
                              hipStream_t stream)
{
    const float* x  = (const float*)d_in[0];   // [8192, 4096] f32
    const float* wg = (const float*)d_in[1];   // [64, 4096]   f32
    float* out = (float*)d_out;                // [1 + 2*S*E*C] f32

    float* wsf      = (float*)d_ws;
    float* gate_top = wsf;                     // [8192] f32
    int*   idx      = (int*)(wsf + 8192);      // [8192] i32
    int*   lrank    = (int*)(wsf + 16384);     // [8192] i32
    int*   hist     = (int*)(wsf + 24576);     // [64*64] i32
    int*   base     = (int*)(wsf + 28672);     // [64*64] i32
    float* me       = wsf + 32768;             // [64] f32

    zero_kernel        <<<8192, 256, 0, stream>>>(out, me);
    gemm_softmax_kernel<<<SDIM / 32, 256, 0, stream>>>(x, wg, gate_top, idx, me);
    rank_kernel        <<<SDIM / 128, 128, 0, stream>>>(idx, lrank, hist);
    prefix_laux_kernel <<<1, 64, 0, stream>>>(hist, base, me, out);
    scatter_kernel     <<<SDIM / 256, 256, 0, stream>>>(idx, lrank, base, gate_top, out);
}